// AvgEncoder_32091995636389
// MI455X (gfx1250) — compile-verified
//
#include <hip/hip_runtime.h>
#include <hip/hip_bf16.h>

// ---------------- problem constants (from reference) ----------------
#define BSZ      64
#define NPB      64
#define ITEMS    (BSZ * NPB)   // 4096 bags
#define PAD      64            // tokens per bag
#define D        128           // embedding dim (floats)
#define ROW_F    D             // floats per embedding row (512 bytes)

// ---------------- kernel tiling ----------------
#define WAVES    4                     // waves per block
#define TPB      (WAVES * 32)          // 128 threads
#define CHUNK    8                     // token rows staged per chunk
#define NCHUNK   (PAD / CHUNK)         // 8 chunks
#define BUF_F    (CHUNK * ROW_F)       // 1024 floats = 4 KB per buffer

// ---------------- CDNA5 async-to-LDS helpers ----------------
typedef int v4i __attribute__((ext_vector_type(4)));
typedef __attribute__((address_space(1))) v4i* gptr_v4i;   // global
typedef __attribute__((address_space(3))) v4i* lptr_v4i;   // LDS

#if __has_builtin(__builtin_amdgcn_global_load_async_to_lds_b128)
#define HAVE_ASYNC_LDS_BUILTIN 1
#endif

// Issue one per-lane 16B gather: copies 16 bytes from global address gsrc to LDS address ldst  (tracked by ASYNCcnt)
__device__ __forceinline__ void async_row16(const float* gsrc, float* ldst) {
#ifdef HAVE_ASYNC_LDS_BUILTIN
  __builtin_amdgcn_global_load_async_to_lds_b128(
      (gptr_v4i)(gsrc),
      (lptr_v4i)(ldst),
      /*offset=*/0, /*cpol=*/0);
#else
  unsigned loff = (unsigned)(unsigned long long)
      (__attribute__((address_space(3))) void*)(ldst);
  asm volatile("global_load_async_to_lds_b128 %0, %1, off"
               :: "v"(loff), "v"(gsrc) : "memory");
#endif
}

template <int N>
__device__ __forceinline__ void wait_async_le() {
#if __has_builtin(__builtin_amdgcn_s_wait_asynccnt)
  __builtin_amdgcn_s_wait_asynccnt(N);
#else
  asm volatile("s_wait_asynccnt %0" :: "i"(N) : "memory");
#endif
  asm volatile("" ::: "memory");   // keep LDS reads after the wait
}

__device__ __forceinline__ void wait_ds0() {
#if __has_builtin(__builtin_amdgcn_s_wait_dscnt)
  __builtin_amdgcn_s_wait_dscnt(0);
#else
  asm volatile("s_wait_dscnt 0" ::: "memory");
#endif
  asm volatile("" ::: "memory");   // keep async issues after the wait
}

// ---------------- kernel ----------------
__global__ __launch_bounds__(TPB)
void avg_encoder_kernel(const int* __restrict__ ids,
                        const int* __restrict__ tokens,
                        const float* __restrict__ lens,
                        const float* __restrict__ emb,
                        float* __restrict__ out) {
  __shared__ __align__(16) float smem[WAVES * 2 * BUF_F];  // 32 KB

  const int lane = threadIdx.x & 31;
  const int wave = threadIdx.x >> 5;
  const int g = blockIdx.x * WAVES + wave;     // bag index [0, 4096)
  if (g >= ITEMS) return;

  const int   item = ids[g];
  const float inv  = 1.0f / lens[item];

  // 64 token ids for this bag: 2 per lane, broadcast later via shuffle.
  const int tbase = item * PAD;
  const int tokA  = tokens[tbase + lane];
  const int tokB  = tokens[tbase + 32 + lane];

  float* buf0 = &smem[wave * 2 * BUF_F];
  float* buf1 = buf0 + BUF_F;

  float4 acc = make_float4(0.f, 0.f, 0.f, 0.f);

  // --- prologue: stage chunk 0 (8 rows, 8 async b128 gathers) ---
  {
#pragma unroll
    for (int r = 0; r < CHUNK; ++r) {
      const int tok = __shfl(tokA, r, 32);                 // rows 0..7
      const float* gsrc = emb + (size_t)tok * D + lane * 4;
      async_row16(gsrc, buf0 + r * ROW_F + lane * 4);
    }
  }

#pragma unroll
  for (int c = 0; c < NCHUNK; ++c) {
    float* cur = (c & 1) ? buf1 : buf0;

    if (c + 1 < NCHUNK) {
      // Issue chunk c+1 into the other buffer (WAR-guard with dscnt first:
      // ASYNCcnt and DScnt are mutually unordered on CDNA5).
      float* nxt = ((c + 1) & 1) ? buf1 : buf0;
      wait_ds0();
#pragma unroll
      for (int r = 0; r < CHUNK; ++r) {
        const int t = (c + 1) * CHUNK + r;
        const int tok = (t < 32) ? __shfl(tokA, t, 32) : __shfl(tokB, t - 32, 32);
        const float* gsrc = emb + (size_t)tok * D + lane * 4;
        async_row16(gsrc, nxt + r * ROW_F + lane * 4);
      }
      // 8 newest (chunk c+1) may remain in flight; chunk c has retired
      // (async loads complete in order).
      wait_async_le<CHUNK>();
    } else {
      wait_async_le<0>();
    }

    // Consume chunk c: 8 ds_load_b128 per lane, accumulate.
#pragma unroll
    for (int r = 0; r < CHUNK; ++r) {
      const float4 v = ((const float4*)(cur + r * ROW_F))[lane];
      acc.x += v.x; acc.y += v.y; acc.z += v.z; acc.w += v.w;
    }
  }

  const float4 o = make_float4(acc.x * inv, acc.y * inv, acc.z * inv, acc.w * inv);
  ((float4*)out)[(size_t)g * (D / 4) + lane] = o;   // coalesced 512B per wave
}

// ---------------- launch ----------------
extern "C" void kernel_launch(void* const* d_in, const int* in_sizes, int n_in,
                              void* d_out, int out_size, void* d_ws, size_t ws_size,
                              hipStream_t stream) {
  (void)in_sizes; (void)n_in; (void)out_size; (void)d_ws; (void)ws_size;
  const int*   ids    = (const int*)d_in[0];    // [64,64] int32
  const int*   tokens = (const int*)d_in[1];    // [100000,64] int32
  const float* lens   = (const float*)d_in[2];  // [100000] f32
  const float* emb    = (const float*)d_in[3];  // [50000,128] f32
  float*       out    = (float*)d_out;          // [64,64,128] f32

  dim3 grid((ITEMS + WAVES - 1) / WAVES);       // 1024 blocks
  dim3 block(TPB);                              // 128 threads = 4 wave32
  avg_encoder_kernel<<<grid, block, 0, stream>>>(ids, tokens, lens, emb, out);
}